// GammatoneFilterBankAnalysis_25769803776159
// MI455X (gfx1250) — compile-verified
//
#include <hip/hip_runtime.h>

typedef float v2f __attribute__((ext_vector_type(2)));
typedef float v8f __attribute__((ext_vector_type(8)));

#define TLEN   65536
#define CHUNK  256
#define NCHUNK (TLEN / CHUNK)   // 256
#define GAMMA  4

struct Cpx { float re, im; };

// -------- Kernel 1: homogeneous responses H_i[t] per channel ------------
// H layout: H[((k*CHUNK + t)*4 + i)*2 + {re,im}]
__global__ void gt_prep_H(const float* __restrict__ a, float* __restrict__ H, int K) {
    int tid = blockIdx.x * blockDim.x + threadIdx.x;
    if (tid >= K * 4) return;
    int k = tid >> 2, i = tid & 3;
    Cpx tp[4];
    for (int j = 0; j < 4; ++j) {
        tp[j].re = a[(k * 5 + 1 + j) * 2 + 0];
        tp[j].im = a[(k * 5 + 1 + j) * 2 + 1];
    }
    Cpx h[4];
    for (int j = 0; j < 4; ++j) { h[j].re = (j == i) ? 1.0f : 0.0f; h[j].im = 0.0f; }
    float* Hk = H + (size_t)k * CHUNK * 8;
    for (int t = 0; t < CHUNK; ++t) {
        Cpx y; y.re = 0.0f; y.im = 0.0f;
        for (int j = 0; j < 4; ++j) {
            y.re -= tp[j].re * h[j].re - tp[j].im * h[j].im;
            y.im -= tp[j].re * h[j].im + tp[j].im * h[j].re;
        }
        Hk[((size_t)t * 4 + i) * 2 + 0] = y.re;
        Hk[((size_t)t * 4 + i) * 2 + 1] = y.im;
        h[3] = h[2]; h[2] = h[1]; h[1] = h[0]; h[0] = y;
    }
}

// -------- Kernel 2: zero-entry-state exit state per (b,k,chunk) ---------
// exitS layout: [((b*K + k)*NCHUNK + c)*8 + i*2 + {re,im}]  (i: y[end-i])
__global__ void gt_exit_states(const float* __restrict__ x, const float* __restrict__ a,
                               float* __restrict__ exitS, int B, int K) {
    int tid = blockIdx.x * blockDim.x + threadIdx.x;
    int total = B * K * NCHUNK;
    if (tid >= total) return;
    int k = tid % K;
    int c = (tid / K) % NCHUNK;
    int b = tid / (K * NCHUNK);
    float gr = a[(k * 5) * 2 + 0], gi = a[(k * 5) * 2 + 1];
    Cpx tp[4];
    for (int j = 0; j < 4; ++j) {
        tp[j].re = a[(k * 5 + 1 + j) * 2 + 0];
        tp[j].im = a[(k * 5 + 1 + j) * 2 + 1];
    }
    Cpx h[4];
    for (int j = 0; j < 4; ++j) { h[j].re = 0.0f; h[j].im = 0.0f; }
    const float* xp = x + (size_t)b * TLEN + (size_t)c * CHUNK;
    for (int t = 0; t < CHUNK; ++t) {
        float xv = xp[t];
        Cpx y; y.re = gr * xv; y.im = gi * xv;
        for (int j = 0; j < 4; ++j) {
            y.re -= tp[j].re * h[j].re - tp[j].im * h[j].im;
            y.im -= tp[j].re * h[j].im + tp[j].im * h[j].re;
        }
        h[3] = h[2]; h[2] = h[1]; h[1] = h[0]; h[0] = y;
    }
    float* e = exitS + (((size_t)b * K + k) * NCHUNK + c) * 8;
    for (int j = 0; j < 4; ++j) { e[j * 2] = h[j].re; e[j * 2 + 1] = h[j].im; }
}

// -------- Kernel 3: sequential scan of chunk entry states ---------------
// entryS layout: [((k*NCHUNK + c)*16 + b)*8 + i*2 + {re,im}]   (16 = batch slot, WMMA B layout)
__global__ void gt_scan(const float* __restrict__ H, const float* __restrict__ exitS,
                        float* __restrict__ entryS, int B, int K) {
    int tid = blockIdx.x * blockDim.x + threadIdx.x;
    if (tid >= B * K) return;
    int k = tid % K;
    int b = tid / K;
    const float* Hk = H + (size_t)k * CHUNK * 8;
    Cpx T[4][4];                        // T[r][i] = H_i[CHUNK-1-r]
    for (int r = 0; r < 4; ++r)
        for (int i = 0; i < 4; ++i) {
            T[r][i].re = Hk[((size_t)(CHUNK - 1 - r) * 4 + i) * 2 + 0];
            T[r][i].im = Hk[((size_t)(CHUNK - 1 - r) * 4 + i) * 2 + 1];
        }
    Cpx s[4];
    for (int i = 0; i < 4; ++i) { s[i].re = 0.0f; s[i].im = 0.0f; }
    for (int c = 0; c < NCHUNK; ++c) {
        float* ent = entryS + (((size_t)k * NCHUNK + c) * 16 + b) * 8;
        for (int i = 0; i < 4; ++i) { ent[i * 2] = s[i].re; ent[i * 2 + 1] = s[i].im; }
        const float* e = exitS + (((size_t)b * K + k) * NCHUNK + c) * 8;
        Cpx ns[4];
        for (int r = 0; r < 4; ++r) {
            Cpx acc; acc.re = e[r * 2]; acc.im = e[r * 2 + 1];
            for (int i = 0; i < 4; ++i) {
                acc.re += T[r][i].re * s[i].re - T[r][i].im * s[i].im;
                acc.im += T[r][i].re * s[i].im + T[r][i].im * s[i].re;
            }
            ns[r] = acc;
        }
        for (int i = 0; i < 4; ++i) s[i] = ns[i];
    }
}

// -------- Kernel 4: recompute y_p + WMMA entry-state correction ---------
// One wave32 per (k, chunk-pair). Lane = (half h, batch b). Requires B==16.
__global__ void __launch_bounds__(32)
gt_final(const float* __restrict__ x, const float* __restrict__ a,
         const float* __restrict__ H, const float* __restrict__ entryS,
         float* __restrict__ out, int K) {
    __shared__ float yt[2 * 16 * 16 * 2];   // [half][t_local][b][re/im]
    int l = threadIdx.x;
    int pair = blockIdx.x % (NCHUNK / 2);
    int k    = blockIdx.x / (NCHUNK / 2);
    int h = l >> 4;          // which chunk of the pair this lane simulates
    int b = l & 15;          // batch; also WMMA N column / A row M
    int c = pair * 2 + h;

    float gr = a[(k * 5) * 2 + 0], gi = a[(k * 5) * 2 + 1];
    Cpx tp[4];
    for (int j = 0; j < 4; ++j) {
        tp[j].re = a[(k * 5 + 1 + j) * 2 + 0];
        tp[j].im = a[(k * 5 + 1 + j) * 2 + 1];
    }

    // WMMA B operand (entry states) for both tiles of the pair.
    // f32 4x16 B layout: VGPR0 = row K=(lanes<16 ? 0 : 2), VGPR1 = row K=+1, col N = lane&15.
    v2f Bre[2], Bim[2], Bneg[2];
    for (int hh = 0; hh < 2; ++hh) {
        const float* ep = entryS + (((size_t)k * NCHUNK + (pair * 2 + hh)) * 16 + b) * 8 + (h ? 4 : 0);
        Bre[hh].x = ep[0]; Bre[hh].y = ep[2];
        Bim[hh].x = ep[1]; Bim[hh].y = ep[3];
        Bneg[hh].x = -ep[1]; Bneg[hh].y = -ep[3];
    }

    Cpx hs[4];
    for (int j = 0; j < 4; ++j) { hs[j].re = 0.0f; hs[j].im = 0.0f; }
    const float* xp = x + (size_t)b * TLEN + (size_t)c * CHUNK;
    const float* Hk = H + (size_t)k * CHUNK * 8;
    float2* outv = (float2*)out;   // complex64 as float2

    for (int g = 0; g < CHUNK / 16; ++g) {
        // 16 sequential recurrence steps (zero-entry particular solution)
        for (int tt = 0; tt < 16; ++tt) {
            float xv = xp[g * 16 + tt];
            Cpx y; y.re = gr * xv; y.im = gi * xv;
            for (int j = 0; j < 4; ++j) {
                y.re -= tp[j].re * hs[j].re - tp[j].im * hs[j].im;
                y.im -= tp[j].re * hs[j].im + tp[j].im * hs[j].re;
            }
            hs[3] = hs[2]; hs[2] = hs[1]; hs[1] = hs[0]; hs[0] = y;
            int idx = ((h * 16 + tt) * 16 + b) * 2;
            yt[idx] = y.re; yt[idx + 1] = y.im;
        }
        // A operand: H rows t0..t0+15. f32 16x4 A layout: M = lane&15,
        // VGPR0 = K=(lanes<16 ? 0 : 2), VGPR1 = K=+1.
        const float* hp = Hk + ((size_t)(g * 16 + b) * 4) * 2 + (h ? 4 : 0);
        v2f Are, Aim;
        Are.x = hp[0]; Are.y = hp[2];
        Aim.x = hp[1]; Aim.y = hp[3];

        for (int hh = 0; hh < 2; ++hh) {
            // C operand from LDS: VGPR j -> M = j + (lane<16 ? 0 : 8), N = lane&15
            v8f Cre, Cim;
            for (int j = 0; j < 8; ++j) {
                int M = j + h * 8;
                int idx = ((hh * 16 + M) * 16 + b) * 2;
                Cre[j] = yt[idx];
                Cim[j] = yt[idx + 1];
            }
            // y += H * s_entry  (complex via 4 real f32 WMMAs, K=4)
            v8f Dre = __builtin_amdgcn_wmma_f32_16x16x4_f32(false, Are, false, Bre[hh], (short)0, Cre, false, false);
            Dre     = __builtin_amdgcn_wmma_f32_16x16x4_f32(false, Aim, false, Bneg[hh], (short)0, Dre, false, false);
            v8f Dim = __builtin_amdgcn_wmma_f32_16x16x4_f32(false, Are, false, Bim[hh], (short)0, Cim, false, false);
            Dim     = __builtin_amdgcn_wmma_f32_16x16x4_f32(false, Aim, false, Bre[hh], (short)0, Dim, false, false);
            // Store: lane owns (b = N, rows M = j + h*8) of chunk pair*2+hh -> 64B contiguous run
            size_t base = ((size_t)b * K + k) * TLEN + (size_t)(pair * 2 + hh) * CHUNK + g * 16 + h * 8;
            for (int j = 0; j < 8; ++j) {
                float2 v; v.x = Dre[j]; v.y = Dim[j];
                outv[base + j] = v;
            }
        }
    }
}

extern "C" void kernel_launch(void* const* d_in, const int* in_sizes, int n_in,
                              void* d_out, int out_size, void* d_ws, size_t ws_size,
                              hipStream_t stream) {
    (void)n_in; (void)ws_size;
    const float* x = (const float*)d_in[0];   // (B, T) float32
    const float* a = (const float*)d_in[1];   // (K, 5) complex64 viewed as float pairs
    float* out = (float*)d_out;               // (B, K, T) complex64 viewed as float pairs
    (void)out_size;

    int B = in_sizes[0] / TLEN;
    int K = (in_sizes[1] % 10 == 0) ? (in_sizes[1] / 10) : (in_sizes[1] / 5);

    float* w = (float*)d_ws;
    float* H      = w;                                        // K*CHUNK*8 floats
    float* exitS  = H + (size_t)K * CHUNK * 8;                // B*K*NCHUNK*8 floats
    float* entryS = exitS + (size_t)B * K * NCHUNK * 8;       // K*NCHUNK*16*8 floats

    int nPrep = K * 4;
    gt_prep_H<<<(nPrep + 127) / 128, 128, 0, stream>>>(a, H, K);

    int nExit = B * K * NCHUNK;
    gt_exit_states<<<(nExit + 255) / 256, 256, 0, stream>>>(x, a, exitS, B, K);

    int nScan = B * K;
    gt_scan<<<(nScan + 63) / 64, 64, 0, stream>>>(H, exitS, entryS, B, K);

    gt_final<<<K * (NCHUNK / 2), 32, 0, stream>>>(x, a, H, entryS, out, K);
}